// AttentionBiLSTM_Additive_52707838657299
// MI455X (gfx1250) — compile-verified
//
#include <hip/hip_runtime.h>
#include <cstdint>
#include <cstddef>

// ---------------------------------------------------------------------------
// Problem constants (from reference)
// ---------------------------------------------------------------------------
#define NE 300      // N_EMBED
#define NH 256      // N_HIDDEN
#define NG 1024     // 4*N_HIDDEN
#define NB 64       // BATCH
#define NS 512      // SEQ_LEN

typedef __attribute__((ext_vector_type(2))) float v2f;
typedef __attribute__((ext_vector_type(8))) float v8f;

// D = A(16x4,f32) * B(4x16,f32) + C(16x16,f32)  -- native f32 WMMA on CDNA5
#define WMMA_F32(a, b, c) \
  __builtin_amdgcn_wmma_f32_16x16x4_f32(false, (a), false, (b), (short)0, (c), false, false)

__device__ __forceinline__ float fsigmoid(float x) {
  return 1.0f / (1.0f + __expf(-x));
}
__device__ __forceinline__ float ftanh(float x) {
  float e = __expf(-2.0f * fabsf(x));
  float t = (1.0f - e) / (1.0f + e);
  return copysignf(t, x);
}

// ---------------------------------------------------------------------------
// Kernel 1: xg = gather(emb) @ Wi + bi   for both directions (blockIdx.z)
// M = S*B = 32768 rows (r = s*64+b), K = 300, N = 1024.
// Block: 128 threads (4 waves). Each block: one 16-row tile x 128 columns.
// A tile staged in LDS (padded even stride -> aligned float2 frags).
// ---------------------------------------------------------------------------
#define XSTR 302
__global__ __launch_bounds__(128) void embed_gemm_kernel(
    const int* __restrict__ words, const float* __restrict__ emb,
    const float* __restrict__ Wi_f, const float* __restrict__ bi_f,
    const float* __restrict__ Wi_b, const float* __restrict__ bi_b,
    float* __restrict__ xg_f, float* __restrict__ xg_b)
{
  const float* Wi = blockIdx.z ? Wi_b : Wi_f;
  const float* bi = blockIdx.z ? bi_b : bi_f;
  float*       xg = blockIdx.z ? xg_b : xg_f;

  __shared__ __align__(16) float xt[16 * XSTR];

  const int tid = threadIdx.x;
  const int rt  = blockIdx.x;                    // 16-row tile of S*B

  // gather 16 embedding rows into LDS (coalesced along k)
  for (int idx = tid; idx < 16 * NE; idx += 128) {
    int m = idx / NE;
    int k = idx - m * NE;
    int r = rt * 16 + m;
    int s = r >> 6;                              // r = s*64 + b
    int b = r & 63;
    int tok = words[b * NS + s];
    xt[m * XSTR + k] = emb[(size_t)tok * NE + k];
  }
  __syncthreads();

  const int wave  = tid >> 5, lane = tid & 31;
  const int nlo   = lane & 15;
  const int m16   = lane & 15;
  const int khalf = (lane >> 4) << 1;            // 0 or 2
  const int mhi   = (lane >> 4) << 3;            // 0 or 8 (C/D rows)
  const int n0    = blockIdx.y * 128 + wave * 32;

  v8f c0 = {0.f,0.f,0.f,0.f,0.f,0.f,0.f,0.f};
  v8f c1 = {0.f,0.f,0.f,0.f,0.f,0.f,0.f,0.f};

  for (int kb = 0; kb < NE; kb += 4) {
    int kk = kb + khalf;
    v2f a = *(const v2f*)&xt[m16 * XSTR + kk];
    v2f b0, b1;
    b0.x = Wi[(size_t)kk       * NG + n0 + nlo];
    b0.y = Wi[(size_t)(kk + 1) * NG + n0 + nlo];
    b1.x = Wi[(size_t)kk       * NG + n0 + 16 + nlo];
    b1.y = Wi[(size_t)(kk + 1) * NG + n0 + 16 + nlo];
    c0 = WMMA_F32(a, b0, c0);
    c1 = WMMA_F32(a, b1, c1);
  }

  const float bias0 = bi[n0 + nlo];
  const float bias1 = bi[n0 + 16 + nlo];
#pragma unroll
  for (int i = 0; i < 8; ++i) {
    size_t row = (size_t)rt * 16 + i + mhi;
    xg[row * NG + n0 + nlo]      = c0[i] + bias0;
    xg[row * NG + n0 + 16 + nlo] = c1[i] + bias1;
  }
}

// ---------------------------------------------------------------------------
// Kernel 2: persistent forward LSTM.
// 64 workgroups: (mt = batch tile 0..3) x (jc = h-column tile 0..15).
// 4 waves per block, wave g owns gate g (r,f,c_hat,o) -> one 16x16 tile.
// Wh_f slice (256K x 64 cols) cached in LDS for all 512 steps.
// Cell state lives in LDS. h double-buffered in global, grid barrier per step.
// ---------------------------------------------------------------------------
#define WSTR 258
__global__ __launch_bounds__(128) void lstm_fwd_kernel(
    const float* __restrict__ xg, const float* __restrict__ Wh,
    const float* __restrict__ bh, float* __restrict__ hs,
    float* hbuf, unsigned* cnt)
{
  __shared__ __align__(16) float whB[4 * 16 * WSTR];   // col-major [(g*16+n)][k]
  __shared__ __align__(16) float hA[16 * WSTR];        // [m][k]
  __shared__ float gact[4][16][17];
  __shared__ float cst[16][17];

  const int tid   = threadIdx.x;
  const int wave  = tid >> 5, lane = tid & 31;
  const int mt    = blockIdx.x & 3;                    // batch tile
  const int jc    = blockIdx.x >> 2;                   // h-column tile
  const int nlo   = lane & 15;
  const int m16   = lane & 15;
  const int khalf = (lane >> 4) << 1;
  const int mhi   = (lane >> 4) << 3;
  const int g     = wave;                              // gate id
  const int ncol  = g * NH + jc * 16 + nlo;            // column in 1024-gate space

  // cache this block's Wh_f slice (once, reused for 512 steps)
  for (int idx = tid; idx < NH * 64; idx += 128) {
    int k = idx >> 6;
    int cg = idx & 63;
    int gg = cg >> 4, n = cg & 15;
    whB[(gg * 16 + n) * WSTR + k] = Wh[(size_t)k * NG + gg * NH + jc * 16 + n];
  }
  for (int e = tid; e < 256; e += 128) cst[e >> 4][e & 15] = 0.0f;
  const float bias = bh[ncol];
  __syncthreads();

  for (int s = 0; s < NS; ++s) {
    // stage h_prev rows for this batch tile (hbuf pre-zeroed for s==0)
    const float* hprev = hbuf + (s & 1) * (NB * NH);
    for (int idx = tid; idx < 16 * NH; idx += 128) {
      int m = idx >> 8;
      int k = idx & (NH - 1);
      hA[m * WSTR + k] = hprev[(mt * 16 + m) * NH + k];
    }
    __syncthreads();

    // C init = xg_f[s] + bias ; two accumulator chains to hide WMMA latency
    const float* xrow = xg + ((size_t)s * NB) * NG;
    v8f acc0;
    v8f acc1 = {0.f,0.f,0.f,0.f,0.f,0.f,0.f,0.f};
#pragma unroll
    for (int i = 0; i < 8; ++i)
      acc0[i] = xrow[(size_t)(mt * 16 + i + mhi) * NG + ncol] + bias;

#pragma unroll 4
    for (int kb = 0; kb < NH; kb += 8) {
      int kk = kb + khalf;
      v2f a0 = *(const v2f*)&hA[m16 * WSTR + kk];
      v2f b0 = *(const v2f*)&whB[(g * 16 + nlo) * WSTR + kk];
      v2f a1 = *(const v2f*)&hA[m16 * WSTR + kk + 4];
      v2f b1 = *(const v2f*)&whB[(g * 16 + nlo) * WSTR + kk + 4];
      acc0 = WMMA_F32(a0, b0, acc0);
      acc1 = WMMA_F32(a1, b1, acc1);
    }

    // activations per gate, exchange via LDS
#pragma unroll
    for (int i = 0; i < 8; ++i) {
      float v = acc0[i] + acc1[i];
      gact[g][i + mhi][nlo] = (g == 2) ? ftanh(v) : fsigmoid(v);
    }
    __syncthreads();

    // elementwise LSTM update for this (mt, jc) tile
    float* hnext = hbuf + ((s + 1) & 1) * (NB * NH);
    for (int e = tid; e < 256; e += 128) {
      int m = e >> 4, n = e & 15;
      float r  = gact[0][m][n], f = gact[1][m][n];
      float ch = gact[2][m][n], o = gact[3][m][n];
      float c  = f * cst[m][n] + r * ch;
      cst[m][n] = c;
      float h = o * ftanh(c);
      int brow = mt * 16 + m, col = jc * 16 + n;
      hs[((size_t)s * NB + brow) * NH + col] = h;
      hnext[brow * NH + col] = h;
    }

    // device-wide barrier (monotonic counter, counter zeroed per launch)
    __threadfence();
    __syncthreads();
    if (tid == 0) {
      __hip_atomic_fetch_add(cnt, 1u, __ATOMIC_RELEASE, __HIP_MEMORY_SCOPE_AGENT);
      unsigned target = (unsigned)(s + 1) * (unsigned)gridDim.x;
      while (__hip_atomic_load(cnt, __ATOMIC_ACQUIRE, __HIP_MEMORY_SCOPE_AGENT) < target)
        __builtin_amdgcn_s_sleep(2);
    }
    __syncthreads();
    __threadfence();
  }
}

// ---------------------------------------------------------------------------
// Kernel 3: hgate_b = h_final(64x256) @ Wh_b(256x1024) + bh_b   (small WMMA)
// 64 blocks x 4 waves; wave = batch tile, block = column tile.
// ---------------------------------------------------------------------------
__global__ __launch_bounds__(128) void hgate_kernel(
    const float* __restrict__ hfinal, const float* __restrict__ Wh_b,
    const float* __restrict__ bh_b, float* __restrict__ hgate)
{
  const int tid   = threadIdx.x;
  const int wave  = tid >> 5, lane = tid & 31;
  const int mt    = wave;
  const int nt    = blockIdx.x;                        // 0..63
  const int nlo   = lane & 15, m16 = lane & 15;
  const int khalf = (lane >> 4) << 1, mhi = (lane >> 4) << 3;

  v8f c = {0.f,0.f,0.f,0.f,0.f,0.f,0.f,0.f};
  for (int kb = 0; kb < NH; kb += 4) {
    int kk = kb + khalf;
    v2f a, b;
    a.x = hfinal[(mt * 16 + m16) * NH + kk];
    a.y = hfinal[(mt * 16 + m16) * NH + kk + 1];
    b.x = Wh_b[(size_t)kk       * NG + nt * 16 + nlo];
    b.y = Wh_b[(size_t)(kk + 1) * NG + nt * 16 + nlo];
    c = WMMA_F32(a, b, c);
  }
  const float bias = bh_b[nt * 16 + nlo];
#pragma unroll
  for (int i = 0; i < 8; ++i)
    hgate[(size_t)(mt * 16 + i + mhi) * NG + nt * 16 + nlo] = c[i] + bias;
}

// ---------------------------------------------------------------------------
// Kernel 4: backward scan — no hidden recurrence matmul, pure elementwise.
// One thread per (b,h): independent length-512 scan, s = 511..0.
// ---------------------------------------------------------------------------
__global__ __launch_bounds__(256) void lstm_bwd_kernel(
    const float* __restrict__ xg_b, const float* __restrict__ hgate,
    float* __restrict__ hs_b)
{
  const int t = blockIdx.x * 256 + threadIdx.x;        // 0..16383
  const int b = t >> 8, h = t & (NH - 1);

  const float hg_r = hgate[b * NG + h];
  const float hg_f = hgate[b * NG + NH + h];
  const float hg_c = hgate[b * NG + 2 * NH + h];
  const float hg_o = hgate[b * NG + 3 * NH + h];

  float c = 0.0f;
  for (int s = NS - 1; s >= 0; --s) {
    size_t base = ((size_t)s * NB + b) * NG;
    if (s > 0)
      __builtin_prefetch(xg_b + base - (size_t)NB * NG + h, 0, 0);
    float r  = fsigmoid(xg_b[base + h]          + hg_r);
    float f  = fsigmoid(xg_b[base + NH + h]     + hg_f);
    float ch = ftanh  (xg_b[base + 2 * NH + h]  + hg_c);
    float o  = fsigmoid(xg_b[base + 3 * NH + h] + hg_o);
    c = f * c + r * ch;
    hs_b[((size_t)s * NB + b) * NH + h] = o * ftanh(c);
  }
}

// ---------------------------------------------------------------------------
// Kernel 5: out = concat(hs_f, hs_b)(32768 x 512) @ W_fc(512x2) + b_fc
// One wave per row, shuffle reduction.
// ---------------------------------------------------------------------------
__global__ __launch_bounds__(256) void fc_kernel(
    const float* __restrict__ hs_f, const float* __restrict__ hs_b,
    const float* __restrict__ Wfc, const float* __restrict__ bfc,
    float* __restrict__ out)
{
  __shared__ float w[2 * NH * 2];
  __shared__ float bb[2];
  const int tid = threadIdx.x;
  for (int i = tid; i < 2 * NH * 2; i += 256) w[i] = Wfc[i];
  if (tid < 2) bb[tid] = bfc[tid];
  __syncthreads();

  const int wave = tid >> 5, lane = tid & 31;
  const size_t row = (size_t)blockIdx.x * 8 + wave;    // 0..32767

  float a0 = 0.f, a1 = 0.f;
#pragma unroll
  for (int tcnt = 0; tcnt < 8; ++tcnt) {
    int k = lane + tcnt * 32;
    float vf = hs_f[row * NH + k];
    float vb = hs_b[row * NH + k];
    a0 += vf * w[k * 2]     + vb * w[(NH + k) * 2];
    a1 += vf * w[k * 2 + 1] + vb * w[(NH + k) * 2 + 1];
  }
  for (int off = 16; off > 0; off >>= 1) {
    a0 += __shfl_down(a0, off, 32);
    a1 += __shfl_down(a1, off, 32);
  }
  if (lane == 0) {
    out[row * 2]     = a0 + bb[0];
    out[row * 2 + 1] = a1 + bb[1];
  }
}

// ---------------------------------------------------------------------------
// Host-side launcher
// ---------------------------------------------------------------------------
extern "C" void kernel_launch(void* const* d_in, const int* in_sizes, int n_in,
                              void* d_out, int out_size, void* d_ws, size_t ws_size,
                              hipStream_t stream) {
  const int*   words = (const int*)  d_in[0];
  const float* emb   = (const float*)d_in[1];
  const float* Wi_f  = (const float*)d_in[2];
  const float* bi_f  = (const float*)d_in[3];
  const float* Wh_f  = (const float*)d_in[4];
  const float* bh_f  = (const float*)d_in[5];
  const float* Wi_b  = (const float*)d_in[6];
  const float* bi_b  = (const float*)d_in[7];
  const float* Wh_b  = (const float*)d_in[8];
  const float* bh_b  = (const float*)d_in[9];
  const float* W_fc  = (const float*)d_in[10];
  const float* b_fc  = (const float*)d_in[11];
  float* out = (float*)d_out;

  char* ws = (char*)d_ws;
  float*    xg_f  = (float*)(ws + 0ull);                   // 128 MB
  float*    xg_b  = (float*)(ws + 134217728ull);           // 128 MB
  float*    hs_f  = (float*)(ws + 268435456ull);           // 32 MB
  float*    hs_b  = (float*)(ws + 301989888ull);           // 32 MB
  float*    hbuf  = (float*)(ws + 335544320ull);           // 2 x 64KB (h double buffer)
  float*    hgate = (float*)(ws + 335675392ull);           // 256 KB
  unsigned* cnt   = (unsigned*)(ws + 335937536ull);        // barrier counter

  // zero h double-buffer (h0 = 0) and the grid-barrier counter (graph-capturable)
  hipMemsetAsync(hbuf, 0, 2 * NB * NH * sizeof(float), stream);
  hipMemsetAsync(cnt, 0, sizeof(unsigned), stream);

  // 1) xg_f / xg_b : gather + GEMM (z=0 forward, z=1 backward)
  dim3 g1((NS * NB) / 16, NG / 128, 2);
  embed_gemm_kernel<<<g1, 128, 0, stream>>>(words, emb, Wi_f, bi_f, Wi_b, bi_b,
                                            xg_f, xg_b);

  // 2) forward LSTM (persistent, grid barrier per step); h_final -> hbuf[0]
  lstm_fwd_kernel<<<64, 128, 0, stream>>>(xg_f, Wh_f, bh_f, hs_f, hbuf, cnt);

  // 3) hgate_b = h_final @ Wh_b + bh_b
  hgate_kernel<<<64, 128, 0, stream>>>(hbuf, Wh_b, bh_b, hgate);

  // 4) backward elementwise scan
  lstm_bwd_kernel<<<64, 256, 0, stream>>>(xg_b, hgate, hs_b);

  // 5) final FC
  fc_kernel<<<(NS * NB) / 8, 256, 0, stream>>>(hs_f, hs_b, W_fc, b_fc, out);
}